// grid_sample_layer_89180700934392
// MI455X (gfx1250) — compile-verified
//
#include <hip/hip_runtime.h>
#include <math.h>

// ---------------------------------------------------------------------------
// Problem constants (from reference): inputs [2,6,96,96,96] f32,
// ref_img [1,4,160,160,160] f32, offsets [4,3] f32 -> out [2,16,96,96,96] f32
// ---------------------------------------------------------------------------
constexpr int Dd  = 96, Hh = 96, Ww = 96;
constexpr int NSP = Dd * Hh * Ww;          // 884736 spatial points ( = 3456*256 )
constexpr int Dr  = 160, Hr = 160, Wr = 160;
constexpr int VS  = Dr * Hr * Wr;          // 4,096,000 elements per modality
constexpr float INV_PI = 0.31830988618379067154f;

typedef unsigned int v4u __attribute__((ext_vector_type(4)));
typedef int          v8i __attribute__((ext_vector_type(8)));
typedef int          v4i __attribute__((ext_vector_type(4)));
typedef float        f2  __attribute__((ext_vector_type(2)));

// 8-byte load from a 4-byte-aligned address (gfx1250 handles unaligned b64).
__device__ __forceinline__ f2 ld2(const float* p) {
    f2 v;
    __builtin_memcpy(&v, p, sizeof(v));
    return v;
}

#if defined(__AMDGCN__) && __has_builtin(__builtin_amdgcn_tensor_load_to_lds) && \
    __has_builtin(__builtin_amdgcn_s_wait_tensorcnt)
#define USE_TDM 1
#else
#define USE_TDM 0
#endif

__global__ __launch_bounds__(256)
void grid_sample_kernel(const float* __restrict__ inputs,   // [2,6,NSP]
                        const float* __restrict__ vol,      // [4,VS]
                        const float* __restrict__ offsets,  // [4,3]
                        float* __restrict__ out)            // [2,16,NSP]
{
    const int t   = threadIdx.x;
    const int b   = blockIdx.y;
    const int sp0 = blockIdx.x * 256;
    const int sp  = sp0 + t;

    float i0, i1, i2, i3, i4, i5;

#if USE_TDM
    // --- Stage this block's 6 input-channel rows (stride NSP) into LDS via
    // --- the Tensor Data Mover: one 2-D DMA (tile 256 x 6) instead of six
    // --- strided per-lane global loads. Tracked by TENSORcnt.
    __shared__ float sIn[6 * 256];
    if (t == 0) {
        unsigned long long ga =
            (unsigned long long)(const void*)(inputs + (size_t)b * 6 * NSP + sp0);
        unsigned int laddr = (unsigned int)(unsigned long long)(void*)sIn;

        v4u g0;
        g0.x = 1u;                                   // count=1, user descriptor
        g0.y = laddr;                                // lds_addr  (bits 63:32)
        g0.z = (unsigned int)(ga & 0xFFFFFFFFull);   // global_addr[31:0]
        g0.w = (unsigned int)((ga >> 32) & 0x1FFFFFFull) | (2u << 30); // [56:32] + type=2

        v8i g1;
        g1[0] = (int)(2u << 16);       // workgroup_mask=0, data_size=2 (4 bytes)
        g1[1] = (int)(256u << 16);     // tensor_dim0.lo16 = 256 (bits 63:48)
        g1[2] = (int)(6u << 16);       // tensor_dim0.hi=0 | tensor_dim1.lo16 = 6
        g1[3] = (int)(256u << 16);     // tensor_dim1.hi=0 | tile_dim0 = 256
        g1[4] = 6;                     // tile_dim1 = 6, tile_dim2 = 0
        g1[5] = NSP;                   // tensor_dim0_stride.lo32 = 884736
        g1[6] = 0;                     // stride0.hi=0 | tensor_dim1_stride.lo=0
        g1[7] = 0;

        v4i zz4  = {0, 0, 0, 0};                     // groups 2/3 unused (<=2D tensor)
        v8i zz8  = {0, 0, 0, 0, 0, 0, 0, 0};
        // clang-23 / therock-10.0 arity: 6 args (extra int32x8 before cpol)
        __builtin_amdgcn_tensor_load_to_lds(g0, g1, zz4, zz4, zz8, 0);
    }
    __builtin_amdgcn_s_wait_tensorcnt(0);
    __syncthreads();

    i0 = sIn[0 * 256 + t];
    i1 = sIn[1 * 256 + t];
    i2 = sIn[2 * 256 + t];
    i3 = sIn[3 * 256 + t];
    i4 = sIn[4 * 256 + t];
    i5 = sIn[5 * 256 + t];
#else
    // Fallback: direct non-temporal loads (read-once stream; keep L2 for vol).
    const float* ip = inputs + (size_t)b * 6 * NSP + sp;
    i0 = __builtin_nontemporal_load(ip + 0 * (size_t)NSP);
    i1 = __builtin_nontemporal_load(ip + 1 * (size_t)NSP);
    i2 = __builtin_nontemporal_load(ip + 2 * (size_t)NSP);
    i3 = __builtin_nontemporal_load(ip + 3 * (size_t)NSP);
    i4 = __builtin_nontemporal_load(ip + 4 * (size_t)NSP);
    i5 = __builtin_nontemporal_load(ip + 5 * (size_t)NSP);
#endif

    // coords normalized: -1 + 2*((atan2+pi)/(2*pi)) == atan2/pi exactly.
    const float gz = atan2f(-i1, -i0) * INV_PI;   // c0 -> grid z
    const float gy = atan2f(-i3, -i2) * INV_PI;   // c1 -> grid y
    const float gx = atan2f(-i5, -i4) * INV_PI;   // c2 -> grid x

    float* ob = out + (size_t)b * 16 * NSP + sp;

#pragma unroll
    for (int o = 0; o < 4; ++o) {
        // osets = offsets / [Dr, Hr, Wr] applied to (x, y, z) respectively.
        const float ox = offsets[3 * o + 0] * (1.0f / Dr);
        const float oy = offsets[3 * o + 1] * (1.0f / Hr);
        const float oz = offsets[3 * o + 2] * (1.0f / Wr);

        // align_corners=True index mapping
        const float ix = (gx + ox + 1.0f) * (0.5f * (Wr - 1));
        const float iy = (gy + oy + 1.0f) * (0.5f * (Hr - 1));
        const float iz = (gz + oz + 1.0f) * (0.5f * (Dr - 1));

        const float xf = floorf(ix), yf = floorf(iy), zf = floorf(iz);
        const float fx = ix - xf, fy = iy - yf, fz = iz - zf;
        const int   x0 = (int)xf, y0 = (int)yf, z0 = (int)zf;

        const float wx0 = 1.0f - fx, wx1 = fx;
        const float wyv[2] = {1.0f - fy, fy};
        const float wzv[2] = {1.0f - fz, fz};
        const bool  vx0 = (unsigned)x0       < (unsigned)Wr;
        const bool  vx1 = (unsigned)(x0 + 1) < (unsigned)Wr;

        float a0 = 0.f, a1 = 0.f, a2 = 0.f, a3 = 0.f;

#pragma unroll
        for (int c = 0; c < 4; ++c) {           // 4 (z,y) corner rows
            const int   zc  = z0 + (c >> 1);
            const int   yc  = y0 + (c & 1);
            const float wzy = wzv[c >> 1] * wyv[c & 1];
            if (((unsigned)zc < (unsigned)Dr) & ((unsigned)yc < (unsigned)Hr)) {
                const int rowb = (zc * Hr + yc) * Wr;
                if (vx0 & vx1) {
                    // Paired x-corners: one 8B gather per channel (x0 <= 158).
                    const float* p  = vol + rowb + x0;
                    const f2 q0 = ld2(p);
                    const f2 q1 = ld2(p + VS);
                    const f2 q2 = ld2(p + 2 * VS);
                    const f2 q3 = ld2(p + 3 * VS);
                    const float w0 = wzy * wx0, w1 = wzy * wx1;
                    a0 += w0 * q0.x + w1 * q0.y;
                    a1 += w0 * q1.x + w1 * q1.y;
                    a2 += w0 * q2.x + w1 * q2.y;
                    a3 += w0 * q3.x + w1 * q3.y;
                } else if (vx0) {               // only x0 in range
                    const float* p  = vol + rowb + x0;
                    const float w0 = wzy * wx0;
                    a0 += w0 * p[0];      a1 += w0 * p[VS];
                    a2 += w0 * p[2 * VS]; a3 += w0 * p[3 * VS];
                } else if (vx1) {               // only x1 in range
                    const float* p  = vol + rowb + x0 + 1;
                    const float w1 = wzy * wx1;
                    a0 += w1 * p[0];      a1 += w1 * p[VS];
                    a2 += w1 * p[2 * VS]; a3 += w1 * p[3 * VS];
                }
            }
        }

        // out[b][o*4+m][sp]; streamed write-once -> NT stores keep vol in L2.
        float* oc = ob + (size_t)(o * 4) * NSP;
        __builtin_nontemporal_store(a0, oc + 0 * (size_t)NSP);
        __builtin_nontemporal_store(a1, oc + 1 * (size_t)NSP);
        __builtin_nontemporal_store(a2, oc + 2 * (size_t)NSP);
        __builtin_nontemporal_store(a3, oc + 3 * (size_t)NSP);
    }
}

extern "C" void kernel_launch(void* const* d_in, const int* in_sizes, int n_in,
                              void* d_out, int out_size, void* d_ws, size_t ws_size,
                              hipStream_t stream) {
    const float* inputs  = (const float*)d_in[0];   // [2,6,96,96,96]
    const float* ref_img = (const float*)d_in[1];   // [1,4,160,160,160]
    const float* offsets = (const float*)d_in[2];   // [4,3]
    float*       out     = (float*)d_out;           // [2,16,96,96,96]

    dim3 block(256);
    dim3 grid(NSP / 256, 2);                        // 3456 x 2 blocks, 8 waves each
    grid_sample_kernel<<<grid, block, 0, stream>>>(inputs, ref_img, offsets, out);
}